// DiffractiveLayer_pixel_53068615909508
// MI455X (gfx1250) — compile-verified
//
#include <hip/hip_runtime.h>
#include <stdint.h>

typedef __attribute__((ext_vector_type(2))) float v2f;
typedef __attribute__((ext_vector_type(4))) float v4f;
typedef __attribute__((ext_vector_type(8))) float v8f;

#define SZ    200
#define NP    256
#define NB    16
#define SPAD  208
#define SPAD2 (SPAD * SPAD)
#define BLK   10
#define SOUT  (SZ * BLK) /* 2000 */
#define AMPF  6.0f
#define PIF   3.14159265358979323846f

// ---------------------------------------------------------------------------
// 1) DFT matrices: F = exp(-2*pi*i*k*n/200) zero-padded to 208x208,
//    G = conj(F)/200 (inverse incl. 1/N so G..G gives 1/N^2).
// ---------------------------------------------------------------------------
__global__ void init_dft_kernel(float2* __restrict__ Fc, float2* __restrict__ Gc) {
  int idx = blockIdx.x * blockDim.x + threadIdx.x;
  if (idx >= SPAD2) return;
  int k = idx / SPAD, n = idx % SPAD;
  float2 f = make_float2(0.f, 0.f), g = make_float2(0.f, 0.f);
  if (k < SZ && n < SZ) {
    int m = (int)(((long long)k * (long long)n) % SZ);   // keep angle small
    float ang = -2.0f * PIF * (float)m / (float)SZ;
    float c = cosf(ang), s = sinf(ang);
    f.x = c;               f.y = s;
    g.x = c * (1.0f / SZ); g.y = -s * (1.0f / SZ);
  }
  Fc[idx] = f;
  Gc[idx] = g;
}

// ---------------------------------------------------------------------------
// 2) Gumbel-argmax modulation. Forward value of gumbel_softmax_hard is the
//    one-hot of argmax(logits + g) (softmax terms cancel numerically), so we
//    compute phase_func[argmax(v+g1)] and intensity_func[argmax(v+g2)]*6.
//    Deterministic counter-hash RNG (identical bits on every call/replay).
// ---------------------------------------------------------------------------
__device__ __forceinline__ float gumbel_noise(uint32_t idx, uint32_t seed) {
  uint32_t x = idx * 2654435761u + seed;
  x ^= x >> 16; x *= 0x7feb352du;
  x ^= x >> 15; x *= 0x846ca68bu;
  x ^= x >> 16;
  float u = ((float)x + 0.5f) * (1.0f / 4294967296.0f);
  u = fminf(fmaxf(u, 1e-10f), 0.99999988f);
  return -__logf(-__logf(u));
}

__global__ void modulation_kernel(const float* __restrict__ volt,
                                  const float* __restrict__ phase_f,
                                  const float* __restrict__ inten_f,
                                  float2* __restrict__ Mc) {
  int idx = blockIdx.x * blockDim.x + threadIdx.x;  // over SZ*SZ pixels
  if (idx >= SZ * SZ) return;
  const float* v = volt + (size_t)idx * NP;
  float best1 = -1e30f, best2 = -1e30f;
  int p1 = 0, p2 = 0;
  for (int p = 0; p < NP; ++p) {
    float lv = __builtin_nontemporal_load(v + p);   // streamed once, NT hint
    float s1 = lv + gumbel_noise((uint32_t)(idx * NP + p), 0x9E3779B9u);
    float s2 = lv + gumbel_noise((uint32_t)(idx * NP + p), 0x85EBCA6Bu);
    if (s1 > best1) { best1 = s1; p1 = p; }
    if (s2 > best2) { best2 = s2; p2 = p; }
  }
  float ph  = phase_f[p1];
  float amp = inten_f[p2] * AMPF;
  float sp, cp;
  __sincosf(ph, &sp, &cp);
  Mc[idx] = make_float2(amp * cp, amp * sp);
}

// ---------------------------------------------------------------------------
// 3) Pack (B,S,S) real+imag into zero-padded interleaved complex (B,208,208).
// ---------------------------------------------------------------------------
__global__ void pack_kernel(const float* __restrict__ wr,
                            const float* __restrict__ wi,
                            float2* __restrict__ dst) {
  int idx = blockIdx.x * blockDim.x + threadIdx.x;  // over NB*SPAD2
  if (idx >= NB * SPAD2) return;
  int b = idx / SPAD2;
  int rem = idx - b * SPAD2;
  int r = rem / SPAD, c = rem - r * SPAD;
  float2 val = make_float2(0.f, 0.f);
  if (r < SZ && c < SZ) {
    size_t s = ((size_t)b * SZ + r) * SZ + c;
    val.x = wr[s];
    val.y = wi[s];
  }
  dst[idx] = val;
}

// ---------------------------------------------------------------------------
// 4) Complex GEMM (208x208x208) via V_WMMA_F32_16X16X4_F32 (full-fp32 WMMA,
//    no precision loss on the DFT).  One wave per 16x16 complex tile:
//    4 real WMMA chains (rr, ii, ri, ir); Cr = rr - ii, Ci = ri + ir.
//    A frag (16x4 f32): lane L holds row m=L&15; VGPR0/1 = K pair 2*(L>>4).
//    B frag (4x16): lane L holds col n=L&15, same K split.
//    C/D frag: VGPR j -> (M = j + 8*(L>>4), N = L&15).
// ---------------------------------------------------------------------------
__global__ __launch_bounds__(32)
void cgemm_wmma_kernel(float2* __restrict__ Cm,
                       const float2* __restrict__ Am, int batchedA,
                       const float2* __restrict__ Bm, int batchedB) {
  int lane = threadIdx.x;           // 0..31 (wave32)
  int tm = blockIdx.x * 16;
  int tn = blockIdx.y * 16;
  int b  = blockIdx.z;

  const float2* A  = Am + (batchedA ? (size_t)b * SPAD2 : (size_t)0);
  const float2* Bp = Bm + (batchedB ? (size_t)b * SPAD2 : (size_t)0);
  float2* C = Cm + (size_t)b * SPAD2;

  int mr = lane & 15;   // row (A) / col (B,C)
  int kh = lane >> 4;   // which K-pair this half-wave owns

  v8f acc_rr = {}; v8f acc_ii = {}; v8f acc_ri = {}; v8f acc_ir = {};

  const float2* arow = A + (size_t)(tm + mr) * SPAD;

  for (int k = 0; k < SPAD; k += 4) {
    int ka = k + 2 * kh;
    float2 a0 = arow[ka];
    float2 a1 = arow[ka + 1];
    float2 b0 = Bp[(size_t)ka * SPAD + tn + mr];
    float2 b1 = Bp[(size_t)(ka + 1) * SPAD + tn + mr];
    if (k + 8 < SPAD) {                                   // global_prefetch_b8
      __builtin_prefetch(&arow[ka + 8], 0, 1);
      __builtin_prefetch(&Bp[(size_t)(ka + 8) * SPAD + tn + mr], 0, 1);
    }

    v2f ar = {a0.x, a1.x}, ai = {a0.y, a1.y};
    v2f br = {b0.x, b1.x}, bi = {b0.y, b1.y};

    acc_rr = __builtin_amdgcn_wmma_f32_16x16x4_f32(false, ar, false, br, (short)0, acc_rr, false, false);
    acc_ii = __builtin_amdgcn_wmma_f32_16x16x4_f32(false, ai, false, bi, (short)0, acc_ii, false, false);
    acc_ri = __builtin_amdgcn_wmma_f32_16x16x4_f32(false, ar, false, bi, (short)0, acc_ri, false, false);
    acc_ir = __builtin_amdgcn_wmma_f32_16x16x4_f32(false, ai, false, br, (short)0, acc_ir, false, false);
  }

  int n  = tn + mr;
  int mb = tm + kh * 8;
#pragma unroll
  for (int j = 0; j < 8; ++j) {
    float2 o = make_float2(acc_rr[j] - acc_ii[j], acc_ri[j] + acc_ir[j]);
    C[(size_t)(mb + j) * SPAD + n] = o;
  }
}

// ---------------------------------------------------------------------------
// 5) Pointwise multiply spectrum by transfer function h (S x S region only;
//    pad region stays zero because F's padded rows/cols are zero).
// ---------------------------------------------------------------------------
__global__ void hmul_kernel(float2* __restrict__ buf,
                            const float* __restrict__ hr,
                            const float* __restrict__ hi) {
  int idx = blockIdx.x * blockDim.x + threadIdx.x;  // over NB*SZ*SZ
  if (idx >= NB * SZ * SZ) return;
  int b = idx / (SZ * SZ);
  int rc = idx - b * (SZ * SZ);
  int r = rc / SZ, c = rc - r * SZ;
  size_t o = (size_t)b * SPAD2 + (size_t)r * SPAD + c;
  float2 t = buf[o];
  float a = hr[rc], bb = hi[rc];
  buf[o] = make_float2(t.x * a - t.y * bb, t.x * bb + t.y * a);
}

// ---------------------------------------------------------------------------
// 6) Fused modulation + 10x10 pixel expansion.  Two adjacent output complex
//    pixels per thread -> one 16-byte NON-TEMPORAL store (b128, NT hint):
//    the 512 MB output stream must not evict temp/Mc from the 192 MB L2.
//    Adjacent columns 2c,2c+1 always share the same source pixel (BLK=10).
// ---------------------------------------------------------------------------
__global__ void expand_kernel(const float2* __restrict__ temp,
                              const float2* __restrict__ Mc,
                              float* __restrict__ out) {
  long long idx = (long long)blockIdx.x * blockDim.x + threadIdx.x;
  if (idx >= (long long)NB * SOUT * (SOUT / 2)) return;
  int cp = (int)(idx % (SOUT / 2));        // column-pair index
  long long t = idx / (SOUT / 2);
  int row = (int)(t % SOUT);
  int b   = (int)(t / SOUT);
  int col = cp * 2;

  int r = row / BLK, rr = row - r * BLK;
  int c = col / BLK, cc = col - c * BLK;   // cc even in {0,2,4,6,8}

  v4f v = {0.f, 0.f, 0.f, 0.f};
  if (rr >= 1 && rr < 9) {
    float2 x = temp[(size_t)b * SPAD2 + (size_t)r * SPAD + c];
    float2 m = Mc[r * SZ + c];
    float vr = x.x * m.x - x.y * m.y;
    float vi = x.x * m.y + x.y * m.x;
    if (cc >= 1 && cc < 9)     { v.x = vr; v.y = vi; }   // cc in {2,4,6,8}
    if (cc + 1 < 9)            { v.z = vr; v.w = vi; }   // cc+1 in {1,3,5,7}
  }
  __builtin_nontemporal_store(v, (v4f*)(out) + idx);
}

// ---------------------------------------------------------------------------
// Launcher: F/G init -> modulation -> pack -> F*X -> (.)*F -> x h ->
//           G*Y -> (.)*G -> expand.  All on `stream`, graph-capture safe.
// ---------------------------------------------------------------------------
extern "C" void kernel_launch(void* const* d_in, const int* in_sizes, int n_in,
                              void* d_out, int out_size, void* d_ws, size_t ws_size,
                              hipStream_t stream) {
  const float* waves_r = (const float*)d_in[0];
  const float* waves_i = (const float*)d_in[1];
  const float* h_real  = (const float*)d_in[2];
  const float* h_imag  = (const float*)d_in[3];
  const float* voltage = (const float*)d_in[4];
  const float* phase_f = (const float*)d_in[5];
  const float* inten_f = (const float*)d_in[6];

  float2* Fc   = (float2*)d_ws;
  float2* Gc   = Fc + SPAD2;
  float2* Mc   = Gc + SPAD2;
  float2* bufA = Mc + SPAD2;                      // NB * SPAD2 complex
  float2* bufB = bufA + (size_t)NB * SPAD2;       // NB * SPAD2 complex

  init_dft_kernel<<<(SPAD2 + 255) / 256, 256, 0, stream>>>(Fc, Gc);
  modulation_kernel<<<(SZ * SZ + 127) / 128, 128, 0, stream>>>(voltage, phase_f, inten_f, Mc);
  pack_kernel<<<(NB * SPAD2 + 255) / 256, 256, 0, stream>>>(waves_r, waves_i, bufA);

  dim3 gtiles(SPAD / 16, SPAD / 16, NB);          // 13 x 13 x 16, 1 wave/block
  // T1 = F * X
  cgemm_wmma_kernel<<<gtiles, 32, 0, stream>>>(bufB, Fc, 0, bufA, 1);
  // T2 = T1 * F
  cgemm_wmma_kernel<<<gtiles, 32, 0, stream>>>(bufA, bufB, 1, Fc, 0);
  // T2 *= h
  hmul_kernel<<<(NB * SZ * SZ + 255) / 256, 256, 0, stream>>>(bufA, h_real, h_imag);
  // T3 = G * T2
  cgemm_wmma_kernel<<<gtiles, 32, 0, stream>>>(bufB, Gc, 0, bufA, 1);
  // temp = T3 * G
  cgemm_wmma_kernel<<<gtiles, 32, 0, stream>>>(bufA, bufB, 1, Gc, 0);

  long long total = (long long)NB * SOUT * (SOUT / 2);  // 32M column-pairs
  int blocks = (int)((total + 255) / 256);
  expand_kernel<<<blocks, 256, 0, stream>>>(bufA, Mc, (float*)d_out);
}